// Quantum_convolution_36756330119275
// MI455X (gfx1250) — compile-verified
//
#include <hip/hip_runtime.h>

// Quantum convolution, closed form:
//   z_j = cos(pi*phi_j + w_j)
//   out = { z1*z2*z3, z0*z1, z0*z1*z2, z0*z1*z2*z3 }
// Memory-bound (~51 MB @ 23.3 TB/s ~= 2.2 us). CDNA5 path: async global->LDS
// b128 staging (ASYNCcnt) + hardware v_cos_f32.

#define INV_2PI 0.15915494309189535f
#define THREADS 256
#define IMGS_PER_BLOCK 8
#define FLOATS_PER_IMG 784                    // 28*28
#define UNITS_PER_BLOCK (IMGS_PER_BLOCK * 196) // float4 (16B) copy units
#define ITEMS_PER_BLOCK (IMGS_PER_BLOCK * 98)  // each item = 2 adjacent patches

__global__ __launch_bounds__(THREADS)
void quanv_kernel(const float* __restrict__ x,
                  const float* __restrict__ w,
                  float* __restrict__ out,
                  int n_images) {
  __shared__ float smem[IMGS_PER_BLOCK * FLOATS_PER_IMG];

  const int tid  = threadIdx.x;
  const int img0 = blockIdx.x * IMGS_PER_BLOCK;
  const int nimg = min(IMGS_PER_BLOCK, n_images - img0);
  const int units = nimg * 196;
  const int items = nimg * 98;

  // ---- Stage nimg whole images (contiguous) into LDS with CDNA5 async copies.
  {
    // Generic address of a __shared__ object is {SHARED_BASE, lds_offset};
    // low 32 bits are exactly the LDS byte address the async op wants.
    const unsigned lds_base = (unsigned)(unsigned long long)(void*)smem;
    const unsigned long long gbase =
        (unsigned long long)(const void*)(x + (size_t)img0 * FLOATS_PER_IMG);
    for (int u = tid; u < units; u += THREADS) {
      unsigned laddr = lds_base + (unsigned)u * 16u;
      unsigned long long gaddr = gbase + (unsigned long long)u * 16ull;
      asm volatile("global_load_async_to_lds_b128 %0, %1, off"
                   :: "v"(laddr), "v"(gaddr) : "memory");
    }
    asm volatile("s_wait_asynccnt 0x0" ::: "memory");
  }
  __syncthreads();

  // Angles in "revolutions" for v_cos_f32 (hw computes cos(2*pi*x)).
  const float w0 = w[0] * INV_2PI;
  const float w1 = w[1] * INV_2PI;
  const float w2 = w[2] * INV_2PI;
  const float w3 = w[3] * INV_2PI;

  for (int it = tid; it < items; it += THREADS) {
    const int li = it / 98;            // local image
    const int pr = it - li * 98;       // pair index within image
    const int r  = pr / 7;             // patch row 0..13
    const int c  = pr - r * 7;         // patch-pair col 0..6 (covers cols 2c,2c+1)

    // Top row 2r and bottom row 2r+1; 16B-aligned float4 loads from LDS.
    const float* base = &smem[li * FLOATS_PER_IMG + r * 56 + c * 4];
    const float4 top = *(const float4*)(base);        // x[2r,   4c..4c+3]
    const float4 bot = *(const float4*)(base + 28);   // x[2r+1, 4c..4c+3]

    // Patch A = (r, 2c): phi0=top.x phi1=top.y phi2=bot.x phi3=bot.y
    const float za0 = __builtin_amdgcn_cosf(0.5f * top.x + w0);
    const float za1 = __builtin_amdgcn_cosf(0.5f * top.y + w1);
    const float za2 = __builtin_amdgcn_cosf(0.5f * bot.x + w2);
    const float za3 = __builtin_amdgcn_cosf(0.5f * bot.y + w3);
    // Patch B = (r, 2c+1)
    const float zb0 = __builtin_amdgcn_cosf(0.5f * top.z + w0);
    const float zb1 = __builtin_amdgcn_cosf(0.5f * top.w + w1);
    const float zb2 = __builtin_amdgcn_cosf(0.5f * bot.z + w2);
    const float zb3 = __builtin_amdgcn_cosf(0.5f * bot.w + w3);

    const float a01 = za0 * za1, a23 = za2 * za3;
    const float b01 = zb0 * zb1, b23 = zb2 * zb3;
    const float4 pa = make_float4(za1 * a23, a01, a01 * za2, a01 * a23);
    const float4 pb = make_float4(zb1 * b23, b01, b01 * zb2, b01 * b23);

    float* o = out + ((size_t)(img0 + li) * 196 + (size_t)(r * 14 + c * 2)) * 4;
    *(float4*)(o)     = pa;   // patch (r, 2c)   -> <Z0..Z3>
    *(float4*)(o + 4) = pb;   // patch (r, 2c+1) -> <Z0..Z3>
  }
}

extern "C" void kernel_launch(void* const* d_in, const int* in_sizes, int n_in,
                              void* d_out, int out_size, void* d_ws, size_t ws_size,
                              hipStream_t stream) {
  (void)n_in; (void)out_size; (void)d_ws; (void)ws_size;
  const float* x = (const float*)d_in[0];        // [B,28,28,1] fp32
  const float* w = (const float*)d_in[1];        // [1,4] fp32
  float* out = (float*)d_out;                    // [B,14,14,4] fp32

  const int B = in_sizes[0] / FLOATS_PER_IMG;    // 8192
  const int blocks = (B + IMGS_PER_BLOCK - 1) / IMGS_PER_BLOCK;
  quanv_kernel<<<blocks, THREADS, 0, stream>>>(x, w, out, B);
}